// AdaptedEntropyModel_7035156431604
// MI455X (gfx1250) — compile-verified
//
#include <hip/hip_runtime.h>
#include <cfloat>
#include <cstdint>

typedef __attribute__((ext_vector_type(4))) float        f32x4;
typedef __attribute__((ext_vector_type(4))) int          i32x4;
typedef __attribute__((ext_vector_type(4))) unsigned int u32x4;
typedef __attribute__((ext_vector_type(8))) int          i32x8;

#define NLEV 64
#define REP  32   // wave32: one LDS replica column per lane -> conflict-free

// Branchless 6-step counting search over 64 sorted values (last = +inf pad).
// Returns #{ i : mcol[i*REP] < r }  ==  nearest-level index.
__device__ __forceinline__ int bsearch64(const float* __restrict__ mcol, float r) {
  int b = 0;
  b += (mcol[(b + 31) * REP] < r) ? 32 : 0;
  b += (mcol[(b + 15) * REP] < r) ? 16 : 0;
  b += (mcol[(b +  7) * REP] < r) ?  8 : 0;
  b += (mcol[(b +  3) * REP] < r) ?  4 : 0;
  b += (mcol[(b +  1) * REP] < r) ?  2 : 0;
  b += (mcol[(b +  0) * REP] < r) ?  1 : 0;
  return b;
}

__global__ __launch_bounds__(256) void sos_quant_kernel(
    const float* __restrict__ x,
    const float* __restrict__ means,
    const float* __restrict__ codebook,
    int*   __restrict__ sym,
    float* __restrict__ yhat,
    long long n)
{
  __shared__ float cbraw[NLEV];        // raw codebook staged by TDM
  __shared__ float mrep[NLEV * REP];   // midpoints, 32-way replicated
  __shared__ float crep[NLEV * REP];   // levels,    32-way replicated

  const int tid  = threadIdx.x;
  const int lane = tid & (REP - 1);

  // ---- Stage the 64-entry codebook into LDS via the Tensor Data Mover ----
#if __has_builtin(__builtin_amdgcn_tensor_load_to_lds)
  if (tid < 32) {  // wave 0 issues the TDM op (EXEC ignored by TDM)
    uint64_t ga  = (uint64_t)(uintptr_t)codebook;
    // Low 32 bits of a flat LDS pointer are the LDS byte offset (ISA 10.2).
    uint32_t lds = (uint32_t)(uintptr_t)&cbraw[0];
    u32x4 g0;
    g0[0] = 1u;                                                  // count=1, user desc
    g0[1] = lds;                                                 // lds_addr
    g0[2] = (uint32_t)ga;                                        // global_addr[31:0]
    g0[3] = (uint32_t)((ga >> 32) & 0x01FFFFFFu) | (2u << 30);   // addr[56:32] | type=2
    i32x8 g1;
    g1[0] = (int)(2u  << 16);   // data_size = 4B
    g1[1] = (int)(64u << 16);   // tensor_dim0 = 64 (lo16)
    g1[2] = (int)(1u  << 16);   // tensor_dim1 = 1  (lo16)
    g1[3] = (int)(64u << 16);   // tile_dim0 = 64
    g1[4] = 1;                  // tile_dim1 = 1
    g1[5] = 64;                 // tensor_dim0_stride = 64
    g1[6] = (int)(64u << 16);   // tensor_dim1_stride = 64 (lo16)
    g1[7] = 0;
    i32x4 z4 = (i32x4)0;
#if defined(__clang_major__) && (__clang_major__ >= 23)
    i32x8 z8 = (i32x8)0;
    __builtin_amdgcn_tensor_load_to_lds(g0, g1, z4, z4, z8, 0);
#else
    __builtin_amdgcn_tensor_load_to_lds(g0, g1, z4, z4, 0);
#endif
#if __has_builtin(__builtin_amdgcn_s_wait_tensorcnt)
    __builtin_amdgcn_s_wait_tensorcnt(0);
#else
    asm volatile("s_wait_tensorcnt 0x0" ::: "memory");
#endif
  }
#else
  if (tid < NLEV) cbraw[tid] = codebook[tid];
#endif
  __syncthreads();

  // ---- Build 32-way replicated, bank-conflict-free search tables ----
  for (int e = tid; e < NLEV * REP; e += (int)blockDim.x) {
    int   i  = e >> 5;                       // level index, column = e & 31
    float ci = cbraw[i];
    float cn = cbraw[(i + 1) & (NLEV - 1)];
    mrep[e] = (i < NLEV - 1) ? 0.5f * (ci + cn) : FLT_MAX;
    crep[e] = ci;
  }
  __syncthreads();

  const float* __restrict__ mcol = &mrep[lane];
  const float* __restrict__ ccol = &crep[lane];

  const long long n4      = n >> 2;
  const long long gstride = (long long)gridDim.x * blockDim.x;
  const f32x4* __restrict__ x4 = (const f32x4*)x;
  const f32x4* __restrict__ m4 = (const f32x4*)means;
  i32x4* __restrict__ s4 = (i32x4*)sym;
  f32x4* __restrict__ y4 = (f32x4*)yhat;

  for (long long i = (long long)blockIdx.x * blockDim.x + tid; i < n4; i += gstride) {
    __builtin_prefetch(x4 + i + gstride, 0, 0);   // global_prefetch_b8 (speculative OK)
    __builtin_prefetch(m4 + i + gstride, 0, 0);
    f32x4 xv = __builtin_nontemporal_load(x4 + i);
    f32x4 mv = __builtin_nontemporal_load(m4 + i);
    i32x4 so;
    f32x4 yo;
#pragma unroll
    for (int c = 0; c < 4; ++c) {                 // 4 independent searches -> ILP
      float r = xv[c] - mv[c];
      int   s = bsearch64(mcol, r);
      so[c] = s;
      yo[c] = ccol[s * REP] + mv[c];
    }
    __builtin_nontemporal_store(so, s4 + i);
    __builtin_nontemporal_store(yo, y4 + i);
  }

  // scalar tail (n % 4 != 0; not hit for the reference shapes)
  for (long long i = (n4 << 2) + (long long)blockIdx.x * blockDim.x + tid; i < n;
       i += gstride) {
    float r = x[i] - means[i];
    int   s = bsearch64(mcol, r);
    sym[i]  = s;
    yhat[i] = ccol[s * REP] + means[i];
  }
}

extern "C" void kernel_launch(void* const* d_in, const int* in_sizes, int n_in,
                              void* d_out, int out_size, void* d_ws, size_t ws_size,
                              hipStream_t stream) {
  const float* x        = (const float*)d_in[0];
  const float* means    = (const float*)d_in[1];
  const float* codebook = (const float*)d_in[2];
  const long long n = (long long)in_sizes[0];

  int*   sym  = (int*)d_out;            // output 0: int32 symbols (bit pattern)
  float* yhat = (float*)d_out + n;      // output 1: float32 y_hat

  const int threads = 256;              // 8 wave32s / WG
  long long want = ((n >> 2) + threads - 1) / threads;
  if (want < 1) want = 1;
  int blocks = (int)(want < 4096 ? want : 4096);

  hipLaunchKernelGGL(sos_quant_kernel, dim3(blocks), dim3(threads), 0, stream,
                     x, means, codebook, sym, yhat, n);
}